// WeightedSAGEConv_60327110639806
// MI455X (gfx1250) — compile-verified
//
#include <hip/hip_runtime.h>

#define NN 50000      // nodes
#define NE 800000     // edges
#define DI 64         // in feats
#define DT 128        // concat feats
#define DOUT 64       // out feats
#define LDSW 130      // padded LDS row pitch for W (bank-conflict-free)

typedef __attribute__((ext_vector_type(2))) float v2f;
typedef __attribute__((ext_vector_type(8))) float v8f;

// ---------------------------------------------------------------- zero ws ---
__global__ __launch_bounds__(256) void sage_zero(float* ws) {
  int i = blockIdx.x * 256 + threadIdx.x;
  if (i < NN * DI + NN) ws[i] = 0.0f;
}

// ----------------------------------------------- edge scatter (mean parts) ---
// one wave32 per edge; lane handles 2 consecutive features (float2)
__global__ __launch_bounds__(256) void sage_scatter(
    const float* __restrict__ h, const float* __restrict__ w,
    const int* __restrict__ src, const int* __restrict__ dst,
    float* __restrict__ hsum, float* __restrict__ deg) {
  unsigned gid  = blockIdx.x * 256u + threadIdx.x;
  unsigned e    = gid >> 5;
  unsigned lane = gid & 31u;
  if (e >= NE) return;
  int s = src[e];
  int d = dst[e];
  float we = w[e];
  v2f hv = *(const v2f*)(h + (size_t)s * DI + lane * 2);
  float* o = hsum + (size_t)d * DI + lane * 2;
  atomicAdd(o,     hv.x * we);
  atomicAdd(o + 1, hv.y * we);
  if (lane == 0) atomicAdd(deg + d, 1.0f);
}

// ------------------------- fused mean + concat + linear via f32 WMMA GEMM ---
// out[n, o] = sum_k [h | hsum/deg][n, k] * W[o, k] + b[o]
// block = 256 threads = 8 waves; wave -> 16-node strip x 64 outputs
__global__ __launch_bounds__(256) void sage_gemm(
    const float* __restrict__ h, const float* __restrict__ hsum,
    const float* __restrict__ deg, const float* __restrict__ W,
    const float* __restrict__ b, float* __restrict__ out) {
  __shared__ float wlds[DOUT * LDSW];

  const int tid = threadIdx.x;
  // stage W [64 x 128] into LDS with padded pitch
  for (int idx = tid; idx < DOUT * DT; idx += 256) {
    int o = idx >> 7;       // / 128
    int k = idx & 127;
    wlds[o * LDSW + k] = W[idx];
  }
  __syncthreads();

  const int wave  = tid >> 5;
  const int lane  = tid & 31;
  const int row   = lane & 15;          // M row (A) / N col (B,D) within tile
  const int khalf = (lane >> 4) * 2;    // lanes 16-31 hold K+2,K+3
  const int node_base = blockIdx.x * 128 + wave * 16;
  if (node_base >= NN) return;          // wave-granular: EXEC all-1s for WMMA

  const int node = node_base + row;
  const float inv = 1.0f / fmaxf(deg[node], 1.0f);
  const float* hrow = h    + (size_t)node * DI;
  const float* srow = hsum + (size_t)node * DI;

  v8f acc[4] = {};   // 4 N-tiles of 16 -> 64 outputs

  // K = 0..63 : self features h
#pragma unroll
  for (int kt = 0; kt < 16; ++kt) {
    int k = kt * 4 + khalf;
    v2f a = *(const v2f*)(hrow + k);
#pragma unroll
    for (int t = 0; t < 4; ++t) {
      v2f bm = *(const v2f*)(&wlds[(t * 16 + row) * LDSW + k]);
      acc[t] = __builtin_amdgcn_wmma_f32_16x16x4_f32(
          false, a, false, bm, (short)0, acc[t], false, false);
    }
  }
  // K = 64..127 : neighbor mean hsum * inv_deg (fused mean + concat)
#pragma unroll
  for (int kt = 0; kt < 16; ++kt) {
    int k = kt * 4 + khalf;
    v2f a = *(const v2f*)(srow + k);
    a.x *= inv;
    a.y *= inv;
#pragma unroll
    for (int t = 0; t < 4; ++t) {
      v2f bm = *(const v2f*)(&wlds[(t * 16 + row) * LDSW + 64 + k]);
      acc[t] = __builtin_amdgcn_wmma_f32_16x16x4_f32(
          false, a, false, bm, (short)0, acc[t], false, false);
    }
  }

  // D layout: VGPR v -> M = v + 8*(lane>=16); N = t*16 + (lane&15). Fuse bias.
  const int mhalf = (lane >> 4) * 8;
#pragma unroll
  for (int t = 0; t < 4; ++t) {
    float bias = b[t * 16 + row];
#pragma unroll
    for (int v = 0; v < 8; ++v) {
      out[(size_t)(node_base + mhalf + v) * DOUT + t * 16 + row] = acc[t][v] + bias;
    }
  }
}

// ---------------------------------------------------------------- launcher ---
extern "C" void kernel_launch(void* const* d_in, const int* in_sizes, int n_in,
                              void* d_out, int out_size, void* d_ws, size_t ws_size,
                              hipStream_t stream) {
  const float* h   = (const float*)d_in[0];
  const float* w   = (const float*)d_in[1];
  const int*   src = (const int*)d_in[2];
  const int*   dst = (const int*)d_in[3];
  const float* W   = (const float*)d_in[4];
  const float* b   = (const float*)d_in[5];
  float* out  = (float*)d_out;
  float* hsum = (float*)d_ws;                  // [NN * DI]
  float* deg  = hsum + (size_t)NN * DI;        // [NN]

  int ztotal = NN * DI + NN;
  sage_zero<<<(ztotal + 255) / 256, 256, 0, stream>>>(hsum);

  long long sthreads = (long long)NE * 32;
  sage_scatter<<<(int)((sthreads + 255) / 256), 256, 0, stream>>>(
      h, w, src, dst, hsum, deg);

  sage_gemm<<<(NN + 127) / 128, 256, 0, stream>>>(h, hsum, deg, W, b, out);
}